// mLSTMCell_47175920779412
// MI455X (gfx1250) — compile-verified
//
#include <hip/hip_runtime.h>

typedef __attribute__((ext_vector_type(2))) float v2f;
typedef __attribute__((ext_vector_type(8))) float v8f;

#define B_SZ    512
#define DIN     1024
#define UNITS_N 1024
#define NHEAD   8
#define HDIM    128      // K == V == 128
#define P_TOT   4112     // 2*H*K + H*V + 2*H + UNITS
#define MTILES  4        // 4 x 16 = 64 output rows per wave

// ---------------------------------------------------------------------------
// Kernel 1: proj = [X | Hprev] @ [W ; R] + bias   via V_WMMA_F32_16X16X4_F32
// One wave per 64x16 output strip: 4 m-tiles share each B fragment, so per
// k-step we issue 4 A b64 loads + 2 B b32 loads for 4 WMMAs (1.5 vmem/wmma).
// A-fragment (16x4 f32): lane%16 = M row, lane/16 selects K pair {2h, 2h+1}.
// B-fragment (4x16 f32): mirrored (n,k) mapping, n = lane%16 (column-major B).
// D/C (16x16 f32, 8 VGPRs): row m = 8*(lane/16) + r, col n = lane%16.
// ---------------------------------------------------------------------------
__device__ __forceinline__ void gemm_kslab(
    const float* __restrict__ A,   // 512 x 1024 activations
    const float* __restrict__ Bm,  // 1024 x P weights
    int m0, int n, int half, int l16, v8f acc[MTILES])
{
    #pragma unroll 2
    for (int kk = 0; kk < DIN; kk += 4) {
        const int ka = kk + 2 * half;
        v2f b;
        b.x = Bm[(size_t)(ka + 0) * P_TOT + n];
        b.y = Bm[(size_t)(ka + 1) * P_TOT + n];
        #pragma unroll
        for (int t = 0; t < MTILES; ++t) {
            const float* ar = A + (size_t)(m0 + 16 * t + l16) * DIN + ka;
            v2f a;
            a.x = ar[0];
            a.y = ar[1];
            acc[t] = __builtin_amdgcn_wmma_f32_16x16x4_f32(
                false, a, false, b, (short)0, acc[t], false, false);
        }
    }
}

__global__ __launch_bounds__(32) void proj_gemm_wmma_f32(
    const float* __restrict__ X,     // B x DIN
    const float* __restrict__ Hp,    // B x UNITS
    const float* __restrict__ W,     // DIN x P
    const float* __restrict__ R,     // UNITS x P
    const float* __restrict__ bias,  // P
    float* __restrict__ proj)        // B x P
{
    const int lane = threadIdx.x;          // 0..31, wave32
    const int half = lane >> 4;
    const int l16  = lane & 15;
    const int m0   = blockIdx.y * (16 * MTILES);  // 64-row strip
    const int n    = blockIdx.x * 16 + l16;       // this lane's output column

    v8f acc[MTILES] = {};

    gemm_kslab(X,  W, m0, n, half, l16, acc);     // input projection
    gemm_kslab(Hp, R, m0, n, half, l16, acc);     // recurrent projection

    const float bn = bias[n];
    #pragma unroll
    for (int t = 0; t < MTILES; ++t) {
        #pragma unroll
        for (int r = 0; r < 8; ++r) {
            proj[(size_t)(m0 + 16 * t + 8 * half + r) * P_TOT + n] = acc[t][r] + bn;
        }
    }
}

// ---------------------------------------------------------------------------
// Kernel 2: per-(b,h) state update — the bandwidth-dominant phase
// (256 MiB read + 256 MiB write of C state => ~23 us floor at 23.3 TB/s).
// 128 threads = 4 waves. Wave w owns rows {w, w+4, ...}; lane owns 4
// consecutive columns, streamed as float4 (global_load/store_b128).
//   C_t[r,c] = f*C[r,c] + i*k[r]*v[c]
//   retr[c]  = sum_r C_t[r,c]*q[r]   (wave partials reduced via LDS)
//   n_t = f*n + i*k;  denom = <n_t,q> + 1e-8;  h[c] = sig(o[c])*retr[c]/denom
// ---------------------------------------------------------------------------
__global__ __launch_bounds__(128) void mlstm_state_update(
    const float* __restrict__ proj,   // B x P
    const float* __restrict__ C_tm1,  // B x H*K*V
    const float* __restrict__ n_tm1,  // B x H*K
    float* __restrict__ h_out,        // B x UNITS
    float* __restrict__ C_out,        // B x H*K*V
    float* __restrict__ n_out)        // B x H*K
{
    __shared__ float  sk[HDIM];
    __shared__ float  sq[HDIM];
    __shared__ float  sred[HDIM];
    __shared__ float4 sr4[4 * 32];    // per-(wave, lane) retr partials

    const int tid  = threadIdx.x;     // 0..127
    const int lane = tid & 31;
    const int wid  = tid >> 5;
    const int b    = blockIdx.x >> 3;
    const int h    = blockIdx.x & 7;

    const float* rp = proj + (size_t)b * P_TOT;

    // --- per-row (k index = tid) values: q, k, n_t, denom partial ---------
    const float qv = rp[h * HDIM + tid];
    const float kv = rp[UNITS_N + h * HDIM + tid];
    const float ig = __expf(rp[3 * UNITS_N + h]);                          // exp input gate
    const float fg = 1.0f / (1.0f + __expf(-rp[3 * UNITS_N + NHEAD + h])); // sigmoid forget

    sk[tid] = kv;
    sq[tid] = qv;

    const size_t nidx = (size_t)b * (NHEAD * HDIM) + h * HDIM + tid;
    const float nt = fg * n_tm1[nidx] + ig * kv;
    n_out[nidx] = nt;
    sred[tid] = nt * qv;
    __syncthreads();

    #pragma unroll
    for (int s = 64; s > 0; s >>= 1) {
        if (tid < s) sred[tid] += sred[tid + s];
        __syncthreads();
    }
    const float denom = sred[0] + 1e-8f;

    // --- per-column values for this lane's 4 columns ----------------------
    const int c4 = 4 * lane;          // first of 4 consecutive columns
    const float4 vv4 = *(const float4*)(rp + 2 * UNITS_N + h * HDIM + c4);

    const size_t cbase = ((size_t)b * NHEAD + h) * (HDIM * HDIM);
    const float4* Cin  = (const float4*)(C_tm1 + cbase);
    float4*       Cout = (float4*)(C_out + cbase);

    float4 iv;                        // i * v_c, loop-invariant
    iv.x = ig * vv4.x; iv.y = ig * vv4.y; iv.z = ig * vv4.z; iv.w = ig * vv4.w;

    float4 retr = make_float4(0.f, 0.f, 0.f, 0.f);
    #pragma unroll 4
    for (int r = wid; r < HDIM; r += 4) {
        const int idx = r * (HDIM / 4) + lane;   // float4 index
        const float4 co = Cin[idx];
        const float skr = sk[r];
        float4 cn;
        cn.x = fmaf(fg, co.x, iv.x * skr);
        cn.y = fmaf(fg, co.y, iv.y * skr);
        cn.z = fmaf(fg, co.z, iv.z * skr);
        cn.w = fmaf(fg, co.w, iv.w * skr);
        Cout[idx] = cn;
        const float sqr = sq[r];
        retr.x = fmaf(cn.x, sqr, retr.x);
        retr.y = fmaf(cn.y, sqr, retr.y);
        retr.z = fmaf(cn.z, sqr, retr.z);
        retr.w = fmaf(cn.w, sqr, retr.w);
    }

    sr4[wid * 32 + lane] = retr;
    __syncthreads();

    if (wid == 0) {
        const float4 r0 = sr4[lane];
        const float4 r1 = sr4[32 + lane];
        const float4 r2 = sr4[64 + lane];
        const float4 r3 = sr4[96 + lane];
        const float4 og4 = *(const float4*)(rp + 3 * UNITS_N + 2 * NHEAD + h * HDIM + c4);
        const float inv = 1.0f / denom;
        float4 hv;
        hv.x = (1.0f / (1.0f + __expf(-og4.x))) * (r0.x + r1.x + r2.x + r3.x) * inv;
        hv.y = (1.0f / (1.0f + __expf(-og4.y))) * (r0.y + r1.y + r2.y + r3.y) * inv;
        hv.z = (1.0f / (1.0f + __expf(-og4.z))) * (r0.z + r1.z + r2.z + r3.z) * inv;
        hv.w = (1.0f / (1.0f + __expf(-og4.w))) * (r0.w + r1.w + r2.w + r3.w) * inv;
        *(float4*)(h_out + (size_t)b * UNITS_N + h * HDIM + c4) = hv;
    }
}

// ---------------------------------------------------------------------------
extern "C" void kernel_launch(void* const* d_in, const int* in_sizes, int n_in,
                              void* d_out, int out_size, void* d_ws, size_t ws_size,
                              hipStream_t stream) {
    const float* X    = (const float*)d_in[0];  // inputs           (B, DIN)
    const float* Hp   = (const float*)d_in[1];  // h_tm1            (B, UNITS)
    const float* Cst  = (const float*)d_in[2];  // C_tm1            (B, H*K*V)
    const float* Nst  = (const float*)d_in[3];  // n_tm1            (B, H*K)
    const float* W    = (const float*)d_in[4];  // kernel           (DIN, P)
    const float* R    = (const float*)d_in[5];  // recurrent_kernel (UNITS, P)
    const float* bias = (const float*)d_in[6];  // bias             (P,)

    float* out   = (float*)d_out;
    float* h_out = out;                                           // B*UNITS
    float* C_out = h_out + (size_t)B_SZ * UNITS_N;                // B*H*K*V
    float* n_out = C_out + (size_t)B_SZ * NHEAD * HDIM * HDIM;    // B*H*K

    float* proj = (float*)d_ws;  // needs B*P*4 = 8.4 MB of workspace

    dim3 g1(P_TOT / 16, B_SZ / (16 * MTILES));   // 257 x 8 strips
    proj_gemm_wmma_f32<<<g1, 32, 0, stream>>>(X, Hp, W, R, bias, proj);

    mlstm_state_update<<<B_SZ * NHEAD, HDIM, 0, stream>>>(
        proj, Cst, Nst, h_out, C_out, n_out);
}